// TensorQuantizer_9405978378779
// MI455X (gfx1250) — compile-verified
//
#include <hip/hip_runtime.h>

// Dynamic INT4 block-wise fake quantization (quantize + dequantize), f32.
//   block size B = 128 along last dim, maxbound = 7.0 (signed int4)
//   per block: amax -> scale = amax/7 (or 1 if amax<=0)
//              out = clamp(rint(x / scale), -8, 7) * scale
//
// MI455X design: memory-bound (512 MiB @ 23.3 TB/s ~= 23 us floor).
//   - wave32: one wave == one 128-elem block; lane = one float4 (b128 ld/st)
//   - in-wave amax butterfly via ds_swizzle_b32 + v_max_u32 on abs-bits
//     (non-negative floats order identically to their bit patterns; avoids
//      the v_max_num canonicalization the FP path needs after each swizzle)
//   - scale = amax/7 via true division (bit-faithful dequant factor)
//   - clamp via v_med3_f32; round via v_rndne_f32 (RTE == jnp.round)
//   - non-temporal TH: one-touch stream, don't thrash the 192MB L2

#ifndef FQ_MAXBOUND
#define FQ_MAXBOUND 7.0f
#endif

// native clang vector (NOT HIP_vector_type) so nontemporal builtins accept it
typedef float v4f __attribute__((ext_vector_type(4)));

__device__ __forceinline__ unsigned umax_dev(unsigned a, unsigned b) {
  return a > b ? a : b;   // v_max_u32
}

// wave32 max-reduce + broadcast of a NON-NEGATIVE float, via integer max on
// its bit pattern. imm = (xor<<10) | (or<<5) | and(0x1f), group-of-32 mode.
__device__ __forceinline__ float wave32_absmax_bcast(float m) {
  unsigned b = __float_as_uint(m);
  b = umax_dev(b, (unsigned)__builtin_amdgcn_ds_swizzle((int)b, 0x401f)); // xor 16
  b = umax_dev(b, (unsigned)__builtin_amdgcn_ds_swizzle((int)b, 0x201f)); // xor 8
  b = umax_dev(b, (unsigned)__builtin_amdgcn_ds_swizzle((int)b, 0x101f)); // xor 4
  b = umax_dev(b, (unsigned)__builtin_amdgcn_ds_swizzle((int)b, 0x081f)); // xor 2
  b = umax_dev(b, (unsigned)__builtin_amdgcn_ds_swizzle((int)b, 0x041f)); // xor 1
  return __uint_as_float(b);
}

__global__ __launch_bounds__(256) void fakequant_int4_b128_kernel(
    const float* __restrict__ x, float* __restrict__ out, int nblocks) {
  const int wave = blockIdx.x * (blockDim.x >> 5) + (threadIdx.x >> 5);
  const int lane = threadIdx.x & 31;
  if (wave >= nblocks) return;

  // float4 index: each wave covers 32 float4 = 128 floats = one quant block
  const long long v4idx = (long long)wave * 32 + lane;

  const v4f* __restrict__ xin = reinterpret_cast<const v4f*>(x);
  v4f* __restrict__ xout      = reinterpret_cast<v4f*>(out);

  // single b128 load per lane, non-temporal (one-touch stream)
  v4f v = __builtin_nontemporal_load(&xin[v4idx]);

  // lane-local amax (|.| folds into v_max3/v_max input modifiers)
  float m = fmaxf(fmaxf(fabsf(v.x), fabsf(v.y)),
                  fmaxf(fabsf(v.z), fabsf(v.w)));

  // wave-wide amax, broadcast to all 32 lanes (5 x ds_swizzle_b32 + v_max_u32)
  m = wave32_absmax_bcast(m);

  // scale = amax/7 exactly as the reference computes it; inv = 7/amax for the
  // quantize multiply. One-time per block, amortized over 128 elements.
  const bool  pos   = (m > 0.0f);
  const float scale = pos ? (m / FQ_MAXBOUND) : 1.0f;
  const float inv   = pos ? (FQ_MAXBOUND / m) : 1.0f;

  // per element: mul, rndne (RTE == jnp.round), med3 clamp, mul
  v4f r;
  r.x = __builtin_amdgcn_fmed3f(rintf(v.x * inv), -8.0f, 7.0f) * scale;
  r.y = __builtin_amdgcn_fmed3f(rintf(v.y * inv), -8.0f, 7.0f) * scale;
  r.z = __builtin_amdgcn_fmed3f(rintf(v.z * inv), -8.0f, 7.0f) * scale;
  r.w = __builtin_amdgcn_fmed3f(rintf(v.w * inv), -8.0f, 7.0f) * scale;

  // single b128 non-temporal store per lane
  __builtin_nontemporal_store(r, &xout[v4idx]);
}

extern "C" void kernel_launch(void* const* d_in, const int* in_sizes, int n_in,
                              void* d_out, int out_size, void* d_ws, size_t ws_size,
                              hipStream_t stream) {
  (void)n_in; (void)d_ws; (void)ws_size; (void)out_size;

  const float* x = reinterpret_cast<const float*>(d_in[0]);
  float* out     = reinterpret_cast<float*>(d_out);

  const long long n       = (long long)in_sizes[0];   // 8192*8192
  const int       nblocks = (int)(n / 128);           // one wave per 128-elem block

  const int threads       = 256;                      // 8 waves per workgroup (wave32)
  const int waves_per_tb  = threads / 32;
  const int grid          = (nblocks + waves_per_tb - 1) / waves_per_tb;

  fakequant_int4_b128_kernel<<<grid, threads, 0, stream>>>(x, out, nblocks);
}